// TopK_69630009803092
// MI455X (gfx1250) — compile-verified
//
#include <hip/hip_runtime.h>
#include <cstdint>

// Top-K(64) + ReLU + scatter for 4096 x 24576 fp32, MI455X (gfx1250).
//
// One workgroup per row. The 96KB row is staged to LDS with CDNA5 async
// global->LDS DMA, pulled once into registers (24 keys/lane), then:
//   pass 1: per-wave 256-bin histogram of the key top byte (registers -> LDS)
//   compact: keys in the selected bin appended to a small LDS candidate list
//   passes 2-4: 8-bit radix select over the ~hundreds of candidates
//   scatter: output reconstructed from register keys, nontemporal b128 stores
// Exactly one HBM read and one HBM write -> ~806MB @ 23.3TB/s ~= 35us bound.

constexpr int NCOLS  = 24576;
constexpr int NROWS  = 4096;
constexpr int KTOP   = 64;
constexpr int BLOCK  = 1024;                   // 32 waves (wave32)
constexpr int NWAVES = BLOCK / 32;
constexpr int CPT    = NCOLS / (BLOCK * 4);    // 6 float4 chunks per thread
constexpr int SROW_BYTES  = NCOLS * 4;         // 98304 B staging (reused)
constexpr int WHIST_WORDS = NWAVES * 256;      // 32KB, overlays staging
constexpr int CAND_MAX    = 4096;              // 16KB, overlays staging
constexpr int SMEM_BYTES  = SROW_BYTES + 32;   // + control words

typedef float    v4f __attribute__((ext_vector_type(4)));
typedef uint32_t v4u __attribute__((ext_vector_type(4)));

// Monotonic float->u32 key: descending float order == descending unsigned key.
__device__ __forceinline__ uint32_t f2key(float f) {
  uint32_t u = __float_as_uint(f);
  uint32_t m = (uint32_t)((int32_t)u >> 31);
  return u ^ (m | 0x80000000u);
}
// relu(value) recovered from key: MSB set => value >= +0.0, else relu -> 0.
__device__ __forceinline__ float key2relu(uint32_t key) {
  return (key & 0x80000000u) ? __uint_as_float(key ^ 0x80000000u) : 0.0f;
}

__global__ __launch_bounds__(BLOCK, 1) void topk_scatter_kernel(
    const float* __restrict__ x, float* __restrict__ out) {
  extern __shared__ char smem[];
  uint32_t* whist = reinterpret_cast<uint32_t*>(smem);                    // 32KB
  uint32_t* cand  = reinterpret_cast<uint32_t*>(smem + WHIST_WORDS * 4);  // 16KB
  uint32_t* ctrl  = reinterpret_cast<uint32_t*>(smem + SROW_BYTES);
  // ctrl[0]=prefix  ctrl[1]=remaining rank  ctrl[2]=tie counter
  // ctrl[3]=selected-bin count  ctrl[4]=candidate append counter

  const int tid = threadIdx.x;
  const int wid = tid >> 5;
  const int row = blockIdx.x;
  const float* __restrict__ rowp = x + (size_t)row * NCOLS;

  // ---- Stage the row into LDS via CDNA5 async global->LDS DMA ------------
#pragma unroll
  for (int j = 0; j < CPT; ++j) {
    const int c = tid + j * BLOCK;                       // float4 chunk id
    uint32_t lds_off = (uint32_t)(uintptr_t)(const void*)(smem + 16 * c);
    const void* gptr = (const void*)(rowp + 4 * c);
    asm volatile("global_load_async_to_lds_b128 %0, %1, off"
                 :: "v"(lds_off), "v"(gptr)
                 : "memory");
  }
  asm volatile("s_wait_asynccnt 0x0" ::: "memory");      // my lanes' chunks ready

  // ---- Pull my 24 elements into registers as keys ------------------------
  v4u k[CPT];
#pragma unroll
  for (int j = 0; j < CPT; ++j) {
    v4f v = reinterpret_cast<const v4f*>(smem)[tid + j * BLOCK];
    v4u kk;
#pragma unroll
    for (int q = 0; q < 4; ++q) kk[q] = f2key(v[q]);
    k[j] = kk;
  }
  if (tid == 0) { ctrl[0] = 0u; ctrl[1] = (uint32_t)KTOP; ctrl[2] = 0u; ctrl[4] = 0u; }
  __syncthreads();   // staging fully consumed; whist/cand may now overwrite it

  // ---- Pass 1: per-wave histogram of the key top byte --------------------
  for (int i = tid; i < WHIST_WORDS; i += BLOCK) whist[i] = 0u;
  __syncthreads();
  uint32_t* myh = whist + (wid << 8);
#pragma unroll
  for (int j = 0; j < CPT; ++j)
#pragma unroll
    for (int q = 0; q < 4; ++q) atomicAdd(&myh[k[j][q] >> 24], 1u);
  __syncthreads();
  if (tid < 256) {
    uint32_t s = 0;
    for (int w = 0; w < NWAVES; ++w) s += whist[(w << 8) + tid];
    whist[tid] = s;
  }
  __syncthreads();
  if (tid == 0) {
    uint32_t kth = ctrl[1], cum = 0;
    int d = 255;
    for (; d > 0; --d) {
      uint32_t c = whist[d];
      if (cum + c >= kth) break;
      cum += c;
    }
    ctrl[0] = (uint32_t)d << 24;
    ctrl[1] = kth - cum;
    ctrl[3] = whist[d];
  }
  __syncthreads();
  const uint32_t topbin = ctrl[0];
  const uint32_t mcount = ctrl[3];

  if (mcount <= (uint32_t)CAND_MAX) {
    // ---- Compact matching keys (registers -> candidate list) -------------
#pragma unroll
    for (int j = 0; j < CPT; ++j)
#pragma unroll
      for (int q = 0; q < 4; ++q) {
        uint32_t key = k[j][q];
        if ((key & 0xFF000000u) == topbin) cand[atomicAdd(&ctrl[4], 1u)] = key;
      }
    __syncthreads();
    const uint32_t cnt = ctrl[4];
    // ---- Passes 2-4 over the candidates only -----------------------------
    for (int shift = 16; shift >= 0; shift -= 8) {
      if (tid < 256) whist[tid] = 0u;
      __syncthreads();
      const uint32_t pfx = ctrl[0];
      const uint32_t mhi = 0xFFFFFFFFu << (shift + 8);
      for (uint32_t i = tid; i < cnt; i += BLOCK) {
        uint32_t key = cand[i];
        if ((key & mhi) == pfx) atomicAdd(&whist[(key >> shift) & 0xFFu], 1u);
      }
      __syncthreads();
      if (tid == 0) {
        uint32_t kth = ctrl[1], cum = 0;
        int d = 255;
        for (; d > 0; --d) {
          uint32_t c = whist[d];
          if (cum + c >= kth) break;
          cum += c;
        }
        ctrl[0] = pfx | ((uint32_t)d << shift);
        ctrl[1] = kth - cum;
      }
      __syncthreads();
    }
  } else {
    // ---- Fallback: radix passes over register keys (any distribution) ----
    for (int shift = 16; shift >= 0; shift -= 8) {
      for (int i = tid; i < WHIST_WORDS; i += BLOCK) whist[i] = 0u;
      __syncthreads();
      const uint32_t pfx = ctrl[0];
      const uint32_t mhi = 0xFFFFFFFFu << (shift + 8);
#pragma unroll
      for (int j = 0; j < CPT; ++j)
#pragma unroll
        for (int q = 0; q < 4; ++q) {
          uint32_t key = k[j][q];
          if ((key & mhi) == pfx) atomicAdd(&myh[(key >> shift) & 0xFFu], 1u);
        }
      __syncthreads();
      if (tid < 256) {
        uint32_t s = 0;
        for (int w = 0; w < NWAVES; ++w) s += whist[(w << 8) + tid];
        whist[tid] = s;
      }
      __syncthreads();
      if (tid == 0) {
        uint32_t kth = ctrl[1], cum = 0;
        int d = 255;
        for (; d > 0; --d) {
          uint32_t c = whist[d];
          if (cum + c >= kth) break;
          cum += c;
        }
        ctrl[0] = pfx | ((uint32_t)d << shift);
        ctrl[1] = kth - cum;
      }
      __syncthreads();
    }
  }

  const uint32_t thresh = ctrl[0];   // exact K-th largest key
  const uint32_t need   = ctrl[1];   // ties still to admit

  // ---- Scatter from registers: relu(top-k) into zeros, streamed ----------
  float* __restrict__ orow = out + (size_t)row * NCOLS;
#pragma unroll
  for (int j = 0; j < CPT; ++j) {
    v4f o;
#pragma unroll
    for (int q = 0; q < 4; ++q) {
      uint32_t key = k[j][q];
      float r = 0.0f;
      if (key > thresh) {
        r = key2relu(key);
      } else if (key == thresh) {
        if (atomicAdd(&ctrl[2], 1u) < need) r = key2relu(key);
      }
      o[q] = r;
    }
    __builtin_nontemporal_store(o, reinterpret_cast<v4f*>(orow) + (tid + j * BLOCK));
  }
}

extern "C" void kernel_launch(void* const* d_in, const int* in_sizes, int n_in,
                              void* d_out, int out_size, void* d_ws, size_t ws_size,
                              hipStream_t stream) {
  const float* x = (const float*)d_in[0];
  float* out = (float*)d_out;
  (void)hipFuncSetAttribute((const void*)topk_scatter_kernel,
                            hipFuncAttributeMaxDynamicSharedMemorySize,
                            SMEM_BYTES);
  topk_scatter_kernel<<<NROWS, BLOCK, SMEM_BYTES, stream>>>(x, out);
}